// Gcn_Net_1236950581664
// MI455X (gfx1250) — compile-verified
//
#include <hip/hip_runtime.h>

// GCN 2-layer forward for MI455X (gfx1250), wave32, WMMA f32 16x16x4.
// Shapes: N~100000 nodes, E~3.2M edges, F_IN=512, HID=16, NLAB=64.
// Memory-bound (~1.1 GB moved, ~2 GFLOP): fp32 WMMA keeps reference numerics;
// W1/W2 staged in LDS; features stream once as clause'd b64 loads.

typedef float v2f __attribute__((ext_vector_type(2)));
typedef float v8f __attribute__((ext_vector_type(8)));

#define F_IN 512
#define HID  16
#define NLAB 64

// ---------------- utility kernels ----------------

__global__ void zero_kernel(float* __restrict__ p, long long n) {
    long long i = (long long)blockIdx.x * blockDim.x + threadIdx.x;
    long long s = (long long)gridDim.x * blockDim.x;
    for (; i < n; i += s) p[i] = 0.0f;
}

__global__ void degree_kernel(const int* __restrict__ src, const int* __restrict__ dst,
                              float* __restrict__ degOut, float* __restrict__ degIn,
                              long long nE) {
    long long i = (long long)blockIdx.x * blockDim.x + threadIdx.x;
    long long s = (long long)gridDim.x * blockDim.x;
    for (; i < nE; i += s) {
        atomicAdd(&degOut[src[i]], 1.0f);
        atomicAdd(&degIn[dst[i]], 1.0f);
    }
}

// In-place: deg -> rsqrt(max(deg,1))  (applied to both deg arrays contiguously)
__global__ void norm_kernel(float* __restrict__ deg, long long n) {
    long long i = (long long)blockIdx.x * blockDim.x + threadIdx.x;
    long long s = (long long)gridDim.x * blockDim.x;
    for (; i < n; i += s) deg[i] = rsqrtf(fmaxf(deg[i], 1.0f));
}

// agg1 -> h2in = relu(agg1*norm_dst + b1) * norm_src  (layer-1 epilogue + layer-2 pre-scale)
__global__ void post1_kernel(const float* __restrict__ agg, const float* __restrict__ normDst,
                             const float* __restrict__ normSrc, const float* __restrict__ b1,
                             float* __restrict__ h, long long nNodes) {
    long long total = nNodes * HID;
    long long i = (long long)blockIdx.x * blockDim.x + threadIdx.x;
    long long s = (long long)gridDim.x * blockDim.x;
    for (; i < total; i += s) {
        long long n = i >> 4;
        int c = (int)(i & 15);
        float v = fmaf(agg[i], normDst[n], b1[c]);
        h[i] = fmaxf(v, 0.0f) * normSrc[n];
    }
}

// Edge-parallel SpMM: out[dst] += vals[src], 16 features per edge, lane-per-feature.
__global__ void spmm_kernel(const float* __restrict__ vals, const int* __restrict__ src,
                            const int* __restrict__ dst, float* __restrict__ out,
                            long long nE) {
    long long total = nE * HID;
    long long i = (long long)blockIdx.x * blockDim.x + threadIdx.x;
    long long s = (long long)gridDim.x * blockDim.x;
    for (; i < total; i += s) {
        long long e = i >> 4;
        int c = (int)(i & 15);
        long long sn = src[e];
        long long dn = dst[e];
        atomicAdd(&out[dn * HID + c], vals[sn * HID + c]);
    }
}

// ---------------- layer-1 GEMM: H = (X * norm_src) @ W1, [N,512]@[512,16] ----------------
// One wave per 16-row tile. A fragment (32-bit A 16x4 layout): lane m (m<16) holds row m,
// K=k..k+1; lane m+16 holds row m, K=k+2..k+3. B fragments from LDS-resident W1
// (row-major [512,16] == KxN, exactly WMMA B order).
__global__ __launch_bounds__(128) void gemm1_kernel(const float* __restrict__ X,
                                                    const float* __restrict__ W1,
                                                    const float* __restrict__ normSrc,
                                                    float* __restrict__ H,
                                                    int nNodes) {
    __shared__ float w1s[F_IN * HID];   // 32 KB of the WGP's 320 KB LDS
    for (int i = threadIdx.x; i < (F_IN * HID) / 4; i += blockDim.x)
        ((float4*)w1s)[i] = ((const float4*)W1)[i];
    __syncthreads();

    const int lane = threadIdx.x & 31;
    const int wave = threadIdx.x >> 5;
    const int tile = blockIdx.x * 4 + wave;
    const int tiles = (nNodes + 15) >> 4;
    if (tile >= tiles) return;                 // uniform per wave: EXEC stays all-ones

    const int half = lane >> 4;                // 0: lanes 0-15, 1: lanes 16-31
    const int m    = lane & 15;
    const int kofs = half * 2;
    const int rowBase = tile * 16;
    int row = rowBase + m;
    if (row >= nNodes) row = nNodes - 1;       // clamp reads for a ragged last tile
    const float scale = normSrc[row];
    const float* __restrict__ xrow = X + (long long)row * F_IN;

    v8f acc = {};
    for (int k = 0; k < F_IN; k += 4) {
        float2 xv = *(const float2*)(xrow + k + kofs);   // global_load_b64
        v2f a; a.x = xv.x * scale; a.y = xv.y * scale;
        v2f b;
        b.x = w1s[(k + kofs) * HID + m];                 // ds_load_2addr_b32
        b.y = w1s[(k + kofs + 1) * HID + m];
        acc = __builtin_amdgcn_wmma_f32_16x16x4_f32(
            /*neg_a=*/false, a, /*neg_b=*/false, b,
            /*c_mod=*/(short)0, acc, /*reuse_a=*/false, /*reuse_b=*/false);
    }

    // C layout: VGPR i -> row (rowBase + i + half*8), col m.
    // Wave-uniform full-tile fast path: straight coalesced stores, no per-lane EXEC churn.
    float* __restrict__ hout = H + (long long)(rowBase + half * 8) * HID + m;
    if (rowBase + 16 <= nNodes) {
        #pragma unroll
        for (int i = 0; i < 8; ++i) hout[i * HID] = acc[i];
    } else {
        #pragma unroll
        for (int i = 0; i < 8; ++i) {
            int r = rowBase + i + half * 8;
            if (r < nNodes) H[(long long)r * HID + m] = acc[i];
        }
    }
}

// ---------------- layer-2 GEMM: OUT = (AGG * norm_dst) @ W2 + b2, [N,16]@[16,64] ----------------
__global__ __launch_bounds__(128) void gemm2_kernel(const float* __restrict__ AGG,
                                                    const float* __restrict__ W2,
                                                    const float* __restrict__ b2,
                                                    const float* __restrict__ normDst,
                                                    float* __restrict__ OUT,
                                                    int nNodes) {
    __shared__ float w2s[HID * NLAB];   // 4 KB
    __shared__ float b2s[NLAB];
    for (int i = threadIdx.x; i < (HID * NLAB) / 4; i += blockDim.x)
        ((float4*)w2s)[i] = ((const float4*)W2)[i];
    if (threadIdx.x < NLAB) b2s[threadIdx.x] = b2[threadIdx.x];
    __syncthreads();

    const int lane = threadIdx.x & 31;
    const int wave = threadIdx.x >> 5;
    const int tile = blockIdx.x * 4 + wave;
    const int tiles = (nNodes + 15) >> 4;
    if (tile >= tiles) return;

    const int half = lane >> 4;
    const int m    = lane & 15;
    const int kofs = half * 2;
    const int rowBase = tile * 16;
    int row = rowBase + m;
    if (row >= nNodes) row = nNodes - 1;
    const float scale = normDst[row];
    const float* __restrict__ arow = AGG + (long long)row * HID;

    // Bias rides in the accumulator: D = A*B + C handles "+ b2" for free.
    v8f acc[4];
    #pragma unroll
    for (int nt = 0; nt < 4; ++nt) {
        float bias = b2s[nt * 16 + m];
        #pragma unroll
        for (int i = 0; i < 8; ++i) acc[nt][i] = bias;
    }

    #pragma unroll
    for (int kc = 0; kc < HID; kc += 4) {
        float2 av = *(const float2*)(arow + kc + kofs);
        v2f a; a.x = av.x * scale; a.y = av.y * scale;
        #pragma unroll
        for (int nt = 0; nt < 4; ++nt) {
            v2f b;
            b.x = w2s[(kc + kofs) * NLAB + nt * 16 + m];
            b.y = w2s[(kc + kofs + 1) * NLAB + nt * 16 + m];
            acc[nt] = __builtin_amdgcn_wmma_f32_16x16x4_f32(
                false, a, false, b, (short)0, acc[nt], false, false);
        }
    }

    // Fast path: full tile -> 32 straight coalesced global_store_b32.
    float* __restrict__ oout = OUT + (long long)(rowBase + half * 8) * NLAB + m;
    if (rowBase + 16 <= nNodes) {
        #pragma unroll
        for (int nt = 0; nt < 4; ++nt)
            #pragma unroll
            for (int i = 0; i < 8; ++i)
                oout[i * NLAB + nt * 16] = acc[nt][i];
    } else {
        #pragma unroll
        for (int nt = 0; nt < 4; ++nt)
            #pragma unroll
            for (int i = 0; i < 8; ++i) {
                int r = rowBase + i + half * 8;
                if (r < nNodes) OUT[(long long)r * NLAB + nt * 16 + m] = acc[nt][i];
            }
    }
}

// ---------------- host side ----------------

static inline unsigned grid1d(long long work, int tb) {
    long long b = (work + tb - 1) / tb;
    if (b > (1LL << 20)) b = (1LL << 20);   // grid-stride loops cover the rest
    if (b < 1) b = 1;
    return (unsigned)b;
}

extern "C" void kernel_launch(void* const* d_in, const int* in_sizes, int n_in,
                              void* d_out, int out_size, void* d_ws, size_t ws_size,
                              hipStream_t stream) {
    const float* X    = (const float*)d_in[0];
    const float* W1   = (const float*)d_in[1];
    const float* b1   = (const float*)d_in[2];
    const float* W2   = (const float*)d_in[3];
    const float* b2   = (const float*)d_in[4];
    const int*   esrc = (const int*)d_in[5];
    const int*   edst = (const int*)d_in[6];
    float*       out  = (float*)d_out;

    const int       nNodes = in_sizes[0] / F_IN;
    const long long nE     = in_sizes[5];

    // Workspace: [normSrc N][normDst N][bufA N*16][bufB N*16]  (~13.6 MB)
    float* normSrc = (float*)d_ws;            // starts life as deg_out
    float* normDst = normSrc + nNodes;        // starts life as deg_in
    float* bufA    = normDst + nNodes;        // h1, later h2in
    float* bufB    = bufA + (long long)nNodes * HID;   // agg1, later agg2

    const int tb = 256;
    const int tiles = (nNodes + 15) >> 4;
    const unsigned gemmBlocks = (unsigned)((tiles + 3) / 4);

    // 1) degrees -> norms (in place)
    zero_kernel<<<grid1d(2LL * nNodes, tb), tb, 0, stream>>>(normSrc, 2LL * nNodes);
    degree_kernel<<<grid1d(nE, tb), tb, 0, stream>>>(esrc, edst, normSrc, normDst, nE);
    norm_kernel<<<grid1d(2LL * nNodes, tb), tb, 0, stream>>>(normSrc, 2LL * nNodes);

    // 2) layer 1: project (WMMA) -> SpMM -> fused epilogue
    gemm1_kernel<<<gemmBlocks, 128, 0, stream>>>(X, W1, normSrc, bufA, nNodes);
    zero_kernel<<<grid1d((long long)nNodes * HID, tb), tb, 0, stream>>>(bufB, (long long)nNodes * HID);
    spmm_kernel<<<grid1d(nE * HID, tb), tb, 0, stream>>>(bufA, esrc, edst, bufB, nE);
    post1_kernel<<<grid1d((long long)nNodes * HID, tb), tb, 0, stream>>>(bufB, normDst, normSrc, b1, bufA, nNodes);

    // 3) layer 2: SpMM -> project (WMMA) with fused bias + norm_dst
    zero_kernel<<<grid1d((long long)nNodes * HID, tb), tb, 0, stream>>>(bufB, (long long)nNodes * HID);
    spmm_kernel<<<grid1d(nE * HID, tb), tb, 0, stream>>>(bufA, esrc, edst, bufB, nE);
    gemm2_kernel<<<gemmBlocks, 128, 0, stream>>>(bufB, W2, b2, normDst, out, nNodes);
}